// WaveletMLP_5188320493661
// MI455X (gfx1250) — compile-verified
//
#include <hip/hip_runtime.h>

// CDNA5 / gfx1250 fused wavelet-MLP:
//   fwd 3-level Haar DWT (LDS) -> band GEMMs on V_WMMA_F32_16X16X4_F32 -> inv DWT -> store
// One workgroup (8 waves / 256 threads) handles 32 rows of 512 f32 (two 16-row M-tiles).
// B (weight) fragments are loaded once per K-step and reused across both M-tiles.

typedef __attribute__((ext_vector_type(2))) float v2f;
typedef __attribute__((ext_vector_type(8))) float v8f;

#define ROWS 32
#define DLEN 512
#define ST   516   // padded LDS row stride (floats): 516 % 64 = 4 -> conflict-free A-frag loads
#define NTHR 256

__shared__ float bufA[ROWS * ST];   // x tile -> GEMM outputs y
__shared__ float bufB[ROWS * ST];   // DWT bands [a(64)|d2(64)|d1(128)|d0(256)] -> inv scratch

// One 16x16 output tile for two stacked 16-row M-tiles, K compile-time constant.
// A frag (ISA 7.12.2): lane lo = row M, VGPR0/1 hold K = k+2*hi, k+2*hi+1.
// B frag: B[k][n] = W[n][k] -> two consecutive floats of W row (n0+lo).
template<int K>
__device__ __forceinline__ void band_gemm(const float* __restrict__ W,
                                          const float* __restrict__ bias,
                                          int boff, int n0, int lo, int hi)
{
    const float* Ap0 = &bufB[lo * ST + boff + 2 * hi];
    const float* Ap1 = Ap0 + 16 * ST;
    const float* Wp  = W + (size_t)(n0 + lo) * K + 2 * hi;

    v8f acc0 = {0.f, 0.f, 0.f, 0.f, 0.f, 0.f, 0.f, 0.f};
    v8f acc1 = {0.f, 0.f, 0.f, 0.f, 0.f, 0.f, 0.f, 0.f};
    #pragma unroll 8
    for (int k = 0; k < K; k += 4) {
        v2f bf = *(const v2f*)(Wp + k);
        v2f a0 = *(const v2f*)(Ap0 + k);
        v2f a1 = *(const v2f*)(Ap1 + k);
        acc0 = __builtin_amdgcn_wmma_f32_16x16x4_f32(false, a0, false, bf, (short)0, acc0, false, false);
        acc1 = __builtin_amdgcn_wmma_f32_16x16x4_f32(false, a1, false, bf, (short)0, acc1, false, false);
    }

    float bb = bias[n0 + lo];
    #pragma unroll
    for (int r = 0; r < 8; ++r) {
        bufA[(r + 8 * hi) * ST + boff + n0 + lo]      = acc0[r] + bb;
        bufA[(16 + r + 8 * hi) * ST + boff + n0 + lo] = acc1[r] + bb;
    }
}

__global__ __launch_bounds__(NTHR, 1)
void wavelet_mlp_fused(const float* __restrict__ x,
                       const float* __restrict__ Wd0, const float* __restrict__ bd0,
                       const float* __restrict__ Wd1, const float* __restrict__ bd1,
                       const float* __restrict__ Wd2, const float* __restrict__ bd2,
                       const float* __restrict__ Wa,  const float* __restrict__ ba,
                       float* __restrict__ out)
{
    const int   tid  = threadIdx.x;       // 0..255
    const int   wave = tid >> 5;          // 0..7
    const int   lane = tid & 31;
    const int   lo   = lane & 15;
    const int   hi   = lane >> 4;
    const long  rowBase = (long)blockIdx.x * ROWS;
    const float C = 0.70710678118654752440f;

    // ---- load 32x512 f32 tile (contiguous, float4) ----
    {
        const float4* xg = (const float4*)(x + rowBase * DLEN);
        for (int j = tid; j < ROWS * DLEN / 4; j += NTHR) {
            float4 v = xg[j];
            int r = (j * 4) >> 9;
            int c = (j * 4) & (DLEN - 1);
            *(float4*)&bufA[r * ST + c] = v;
        }
    }
    __syncthreads();

    // ---- fwd level 0: bufA(512) -> bufB: a1 [0..255], d0 [256..511] ----
    for (int i = tid; i < ROWS * 256; i += NTHR) {
        int r = i >> 8, k = i & 255;
        float e = bufA[r * ST + 2 * k];
        float o = bufA[r * ST + 2 * k + 1];
        bufB[r * ST + 256 + k] = (e - o) * C;
        bufB[r * ST + k]       = (e + o) * C;
    }
    __syncthreads();

    // ---- fwd level 1 (in-place in bufB, register-staged): a1[0..255] -> a2[0..127], d1[128..255]
    {
        float av[16], dv[16];
        #pragma unroll
        for (int u = 0; u < 16; ++u) {
            int i = tid + u * NTHR;           // 32*128 items
            int r = i >> 7, k = i & 127;
            float e = bufB[r * ST + 2 * k];
            float o = bufB[r * ST + 2 * k + 1];
            dv[u] = (e - o) * C;
            av[u] = (e + o) * C;
        }
        __syncthreads();
        #pragma unroll
        for (int u = 0; u < 16; ++u) {
            int i = tid + u * NTHR;
            int r = i >> 7, k = i & 127;
            bufB[r * ST + 128 + k] = dv[u];
            bufB[r * ST + k]       = av[u];
        }
    }
    __syncthreads();

    // ---- fwd level 2 (in-place, register-staged): a2[0..127] -> a3[0..63], d2[64..127]
    {
        float av[8], dv[8];
        #pragma unroll
        for (int u = 0; u < 8; ++u) {
            int i = tid + u * NTHR;           // 32*64 items
            int r = i >> 6, k = i & 63;
            float e = bufB[r * ST + 2 * k];
            float o = bufB[r * ST + 2 * k + 1];
            dv[u] = (e - o) * C;
            av[u] = (e + o) * C;
        }
        __syncthreads();
        #pragma unroll
        for (int u = 0; u < 8; ++u) {
            int i = tid + u * NTHR;
            int r = i >> 6, k = i & 63;
            bufB[r * ST + 64 + k] = dv[u];
            bufB[r * ST + k]      = av[u];
        }
    }
    __syncthreads();

    // ---- band GEMMs (bands in bufB -> y in bufA), 352 WMMAs per wave ----
    if (wave < 4) band_gemm<64>(Wa,  ba,  0,  wave * 16,        lo, hi);   // y_a
    else          band_gemm<64>(Wd2, bd2, 64, (wave - 4) * 16,  lo, hi);   // y_d2
    band_gemm<128>(Wd1, bd1, 128, wave * 16,       lo, hi);                // y_d1
    band_gemm<256>(Wd0, bd0, 256, wave * 16,       lo, hi);                // y_d0 lo half
    band_gemm<256>(Wd0, bd0, 256, (wave + 8) * 16, lo, hi);                // y_d0 hi half
    __syncthreads();

    // ---- inv level 2: y_a[0..63] + y_d2[64..127] (bufA) -> bufB[0..127] ----
    for (int i = tid; i < ROWS * 64; i += NTHR) {
        int r = i >> 6, j = i & 63;
        float av = bufA[r * ST + j];
        float dv = bufA[r * ST + 64 + j];
        bufB[r * ST + 2 * j]     = (av + dv) * C;
        bufB[r * ST + 2 * j + 1] = (av - dv) * C;
    }
    __syncthreads();
    // ---- inv level 1: bufB[0..127] + y_d1(bufA[128..255]) -> bufB[256..511] ----
    for (int i = tid; i < ROWS * 128; i += NTHR) {
        int r = i >> 7, j = i & 127;
        float av = bufB[r * ST + j];
        float dv = bufA[r * ST + 128 + j];
        bufB[r * ST + 256 + 2 * j]     = (av + dv) * C;
        bufB[r * ST + 256 + 2 * j + 1] = (av - dv) * C;
    }
    __syncthreads();
    // ---- inv level 0: bufB[256..511] + y_d0(bufA[256..511]) -> global (float2) ----
    for (int i = tid; i < ROWS * 256; i += NTHR) {
        int r = i >> 8, j = i & 255;
        float av = bufB[r * ST + 256 + j];
        float dv = bufA[r * ST + 256 + j];
        float2 ov;
        ov.x = (av + dv) * C;
        ov.y = (av - dv) * C;
        *(float2*)&out[(rowBase + r) * DLEN + 2 * j] = ov;
    }
}

extern "C" void kernel_launch(void* const* d_in, const int* in_sizes, int n_in,
                              void* d_out, int out_size, void* d_ws, size_t ws_size,
                              hipStream_t stream) {
    const float* xp  = (const float*)d_in[0];
    const float* Wd0 = (const float*)d_in[1];
    const float* bd0 = (const float*)d_in[2];
    const float* Wd1 = (const float*)d_in[3];
    const float* bd1 = (const float*)d_in[4];
    const float* Wd2 = (const float*)d_in[5];
    const float* bd2 = (const float*)d_in[6];
    const float* Wa  = (const float*)d_in[7];
    const float* ba  = (const float*)d_in[8];
    float* outp = (float*)d_out;

    int M = in_sizes[0] / DLEN;          // 65536 rows
    int blocks = M / ROWS;               // 2048 workgroups
    wavelet_mlp_fused<<<blocks, NTHR, 0, stream>>>(
        xp, Wd0, bd0, Wd1, bd1, Wd2, bd2, Wa, ba, outp);
}